// StructSVMModel_70119636074682
// MI455X (gfx1250) — compile-verified
//
#include <hip/hip_runtime.h>

// Problem constants (match reference)
#define NPIX   (1024 * 1024)
#define NF     256
#define NC     21
#define NCPAD  32
#define NSP    4096
#define NE     16384

typedef float v2f __attribute__((ext_vector_type(2)));
typedef float v8f __attribute__((ext_vector_type(8)));

__device__ __forceinline__ void atomic_add_f32(float* p, float v) {
    __hip_atomic_fetch_add(p, v, __ATOMIC_RELAXED, __HIP_MEMORY_SCOPE_AGENT);
}

// ---------------------------------------------------------------------------
// Kernel 1: zero the accumulators (sums + counts)
// ---------------------------------------------------------------------------
__global__ void zero_ws_kernel(float* __restrict__ ws, int n) {
    int i = blockIdx.x * blockDim.x + threadIdx.x;
    int stride = gridDim.x * blockDim.x;
    for (; i < n; i += stride) ws[i] = 0.0f;
}

// ---------------------------------------------------------------------------
// Kernel 2: zero-pad weights to 32 rows so GEMM B-loads are unconditional
// (kills the per-iteration EXEC save/restore seen in round 1).
// ---------------------------------------------------------------------------
__global__ void pad_weights_kernel(const float* __restrict__ Wn,
                                   const float* __restrict__ We,
                                   float* __restrict__ WnP,   // 32 x NF
                                   float* __restrict__ WeP) { // 32 x 2*NF
    int i = blockIdx.x * blockDim.x + threadIdx.x;
    int stride = gridDim.x * blockDim.x;
    for (int j = i; j < NCPAD * NF; j += stride) {
        int r = j / NF;
        WnP[j] = (r < NC) ? Wn[j] : 0.0f;          // row-major copy, rows >= 21 -> 0
    }
    for (int j = i; j < NCPAD * 2 * NF; j += stride) {
        int r = j / (2 * NF);
        WeP[j] = (r < NC) ? We[j] : 0.0f;
    }
}

// ---------------------------------------------------------------------------
// Kernel 3: segment scatter-add. One wave per pixel: 32 lanes x 8 feats.
// 1 GB streaming read; atomics land in the 4 MB L2-resident accumulator.
// ---------------------------------------------------------------------------
__global__ void scatter_kernel(const float* __restrict__ feats,
                               const int*   __restrict__ sp,
                               float* __restrict__ sums,
                               float* __restrict__ counts) {
    const int lane   = threadIdx.x & 31;
    const int wave   = (blockIdx.x * blockDim.x + threadIdx.x) >> 5;
    const int nwaves = (gridDim.x * blockDim.x) >> 5;

    for (int p = wave; p < NPIX; p += nwaves) {
        const int s = sp[p];
        const float* src = feats + (size_t)p * NF + lane * 8;
        float4 f0 = *(const float4*)(src);
        float4 f1 = *(const float4*)(src + 4);
        float* dst = sums + (size_t)s * NF + lane * 8;
        atomic_add_f32(dst + 0, f0.x);
        atomic_add_f32(dst + 1, f0.y);
        atomic_add_f32(dst + 2, f0.z);
        atomic_add_f32(dst + 3, f0.w);
        atomic_add_f32(dst + 4, f1.x);
        atomic_add_f32(dst + 5, f1.y);
        atomic_add_f32(dst + 6, f1.z);
        atomic_add_f32(dst + 7, f1.w);
        if (lane == 0) atomic_add_f32(counts + s, 1.0f);
    }
}

// ---------------------------------------------------------------------------
// Kernel 4: node_features = sums / max(counts, 1)
// ---------------------------------------------------------------------------
__global__ void finalize_kernel(float* __restrict__ sums,
                                const float* __restrict__ counts) {
    int i = blockIdx.x * blockDim.x + threadIdx.x;
    int stride = gridDim.x * blockDim.x;
    for (; i < NSP * NF; i += stride) {
        float c = counts[i >> 8];              // NF == 256
        sums[i] = sums[i] / fmaxf(c, 1.0f);
    }
}

// ---------------------------------------------------------------------------
// Kernel 5: node potentials = node_features (4096x256) @ WnP^T (256x32)
// One wave computes one 16x16 D tile with V_WMMA_F32_16X16X4_F32.
// A lane layout: row = lane&15, K pair = (lane>>4)*2  -> float2 loads.
// B lane layout: col = lane&15, same K pair           -> float2 from W row.
// ---------------------------------------------------------------------------
__global__ void node_gemm_kernel(const float* __restrict__ nodef,
                                 const float* __restrict__ WnP,
                                 float* __restrict__ out) {
    const int lane  = threadIdx.x & 31;
    const int wtile = (blockIdx.x * blockDim.x + threadIdx.x) >> 5;  // 0..511
    const int m0 = (wtile >> 1) * 16;        // superpixel tile base
    const int n0 = (wtile & 1) * 16;         // class tile base (0 or 16)
    const int r    = lane & 15;
    const int koff = (lane >> 4) * 2;

    const float* arow = nodef + (size_t)(m0 + r) * NF + koff;
    const float* brow = WnP + (size_t)(n0 + r) * NF + koff;   // padded: always valid

    v8f acc = {0.f, 0.f, 0.f, 0.f, 0.f, 0.f, 0.f, 0.f};
#pragma unroll 4
    for (int kb = 0; kb < NF; kb += 4) {
        v2f a = *(const v2f*)(arow + kb);
        v2f b = *(const v2f*)(brow + kb);
        acc = __builtin_amdgcn_wmma_f32_16x16x4_f32(
            false, a, false, b, (short)0, acc, false, false);
    }

    // D layout: VGPR rr, lane l -> M = rr + (l>>4)*8, N = l&15
    const int mm  = m0 + (lane >> 4) * 8;
    const int col = n0 + (lane & 15);
    if (col < NC) {
#pragma unroll
        for (int rr = 0; rr < 8; ++rr)
            out[(size_t)(mm + rr) * NC + col] = acc[rr];
    }
}

// ---------------------------------------------------------------------------
// Kernel 6: edge potentials. edge_feat = concat(node[src], node[dst]) (512),
// times WeP^T (512x32). Two 256-K passes: src rows vs W[:, :256],
// dst rows vs W[:, 256:]. Gathered A rows hit L2-resident node features.
// ---------------------------------------------------------------------------
__global__ void edge_gemm_kernel(const float* __restrict__ nodef,
                                 const int*   __restrict__ edges,
                                 const float* __restrict__ WeP,
                                 float* __restrict__ out) {
    const int lane  = threadIdx.x & 31;
    const int wtile = (blockIdx.x * blockDim.x + threadIdx.x) >> 5;  // 0..2047
    const int m0 = (wtile >> 1) * 16;        // edge tile base
    const int n0 = (wtile & 1) * 16;
    const int r    = lane & 15;
    const int koff = (lane >> 4) * 2;

    const int e   = m0 + r;
    const int src = edges[2 * e];
    const int dst = edges[2 * e + 1];
    const float* asrc = nodef + (size_t)src * NF + koff;
    const float* adst = nodef + (size_t)dst * NF + koff;
    const float* brow = WeP + (size_t)(n0 + r) * (2 * NF) + koff;  // padded: always valid

    v8f acc = {0.f, 0.f, 0.f, 0.f, 0.f, 0.f, 0.f, 0.f};
#pragma unroll 4
    for (int kb = 0; kb < NF; kb += 4) {               // first half: src rows
        v2f a = *(const v2f*)(asrc + kb);
        v2f b = *(const v2f*)(brow + kb);
        acc = __builtin_amdgcn_wmma_f32_16x16x4_f32(
            false, a, false, b, (short)0, acc, false, false);
    }
#pragma unroll 4
    for (int kb = 0; kb < NF; kb += 4) {               // second half: dst rows
        v2f a = *(const v2f*)(adst + kb);
        v2f b = *(const v2f*)(brow + NF + kb);
        acc = __builtin_amdgcn_wmma_f32_16x16x4_f32(
            false, a, false, b, (short)0, acc, false, false);
    }

    const int mm  = m0 + (lane >> 4) * 8;
    const int col = n0 + (lane & 15);
    if (col < NC) {
#pragma unroll
        for (int rr = 0; rr < 8; ++rr)
            out[(size_t)(mm + rr) * NC + col] = acc[rr];
    }
}

// ---------------------------------------------------------------------------
// Launcher
// ---------------------------------------------------------------------------
extern "C" void kernel_launch(void* const* d_in, const int* in_sizes, int n_in,
                              void* d_out, int out_size, void* d_ws, size_t ws_size,
                              hipStream_t stream) {
    const float* feats = (const float*)d_in[0];   // (1,1024,1024,256) f32
    const int*   sp    = (const int*)  d_in[1];   // (1,1024,1024) int
    const int*   edges = (const int*)  d_in[2];   // (16384,2) int
    const float* Wn    = (const float*)d_in[3];   // (21,256) f32
    const float* We    = (const float*)d_in[4];   // (21,512) f32
    float* out = (float*)d_out;                   // 4096*21 node || 16384*21 edge

    // Workspace layout (floats):
    float* sums   = (float*)d_ws;                         // NSP*NF
    float* counts = sums + (size_t)NSP * NF;              // NSP
    float* WnP    = counts + NSP;                         // 32*NF
    float* WeP    = WnP + (size_t)NCPAD * NF;             // 32*2*NF

    zero_ws_kernel<<<256, 256, 0, stream>>>(sums, NSP * NF + NSP);
    pad_weights_kernel<<<96, 256, 0, stream>>>(Wn, We, WnP, WeP);
    scatter_kernel<<<4096, 256, 0, stream>>>(feats, sp, sums, counts);
    finalize_kernel<<<1024, 256, 0, stream>>>(sums, counts);
    node_gemm_kernel<<<64, 256, 0, stream>>>(sums, WnP, out);                     // 512 wave-tiles
    edge_gemm_kernel<<<256, 256, 0, stream>>>(sums, edges, WeP, out + NSP * NC);  // 2048 wave-tiles
}